// Net_60533269070005
// MI455X (gfx1250) — compile-verified
//
#include <hip/hip_runtime.h>
#include <stdint.h>
#include <stddef.h>

// Problem dims (fixed by the reference)
#define B_DIM 1024
#define S_DIM 200000
#define G_DIM 10000

#define GENE_TILE 16                 // genes per workgroup = WMMA N tile
#define KC 64                        // SNP chunk (K) length per iteration
#define KCP 68                       // padded LDS row stride (bank-conflict free)
#define WAVES 4                      // waves per workgroup (wave32)
#define NTILE 2                      // 16-row batch tiles per wave
#define BLOCK_THREADS (WAVES * 32)   // 128
#define ROWS_PER_WG (WAVES * NTILE * 16)  // 128
#define NBY (B_DIM / ROWS_PER_WG)         // 8

typedef __attribute__((ext_vector_type(2))) float v2f;
typedef __attribute__((ext_vector_type(8))) float v8f;
typedef __attribute__((ext_vector_type(4))) unsigned int v4u;
typedef __attribute__((ext_vector_type(4))) int v4i;
typedef __attribute__((ext_vector_type(8))) int v8i;

// ---------------------------------------------------------------------------
// Kernel 1: segment starts. seg_start[g] = lower_bound(gene_ids, g); [G]=S.
// ---------------------------------------------------------------------------
__global__ void seg_offsets_kernel(const int* __restrict__ gene_ids,
                                   int* __restrict__ seg_start) {
  int g = blockIdx.x * blockDim.x + threadIdx.x;
  if (g > G_DIM) return;
  int lo = 0, hi = S_DIM;
  while (lo < hi) {
    int mid = (lo + hi) >> 1;
    if (gene_ids[mid] < g) lo = mid + 1; else hi = mid;
  }
  seg_start[g] = lo;
}

// ---------------------------------------------------------------------------
// Tensor Data Mover: load a 16-row x KC-col f32 tile of x (row stride S_DIM)
// into LDS, padding 4 DWORDs after every 64 DWORDs -> LDS row stride KCP=68.
// ---------------------------------------------------------------------------
#if defined(__gfx1250__) && __has_builtin(__builtin_amdgcn_tensor_load_to_lds)
#define HAVE_TDM 1
__device__ __forceinline__ void tdm_load_x_tile(uint32_t lds_off,
                                                const float* gptr,
                                                unsigned int cols_avail) {
  uint64_t ga = (uint64_t)(uintptr_t)gptr;
  v4u g0;
  g0[0] = 1u;                                   // count=1 (valid), no gather
  g0[1] = lds_off;                              // lds_addr (bytes)
  g0[2] = (uint32_t)ga;                         // global_addr[31:0]
  g0[3] = (uint32_t)(ga >> 32) | (2u << 30);    // global_addr[56:32] | type=2
  v8i g1;
  // pad_amount=4dw(enc 3) | pad_interval=64dw(enc 5) | pad_enable | data_size=4B(enc 2)
  g1[0] = (int)((3u << 25) | (5u << 22) | (1u << 20) | (2u << 16));
  g1[1] = (int)((cols_avail & 0xFFFFu) << 16);                   // tensor_dim0 lo16
  g1[2] = (int)(((cols_avail >> 16) & 0xFFFFu) | (16u << 16));   // dim0 hi16 | tensor_dim1=16
  g1[3] = (int)(((unsigned)KC) << 16);                           // dim1 hi16=0 | tile_dim0=KC
  g1[4] = (int)16;                                               // tile_dim1=16, tile_dim2=0
  g1[5] = (int)(unsigned)S_DIM;                                  // tensor_dim0_stride lo32
  g1[6] = 0;                                                     // stride hi16 | dim1_stride lo16
  g1[7] = 0;
  v4i g2 = {0, 0, 0, 0};
  v4i g3 = {0, 0, 0, 0};
#if __has_include(<hip/amd_detail/amd_gfx1250_TDM.h>)
  // amdgpu-toolchain (clang-23 / therock-10.0 headers): 6-arg builtin
  v8i g4 = {0, 0, 0, 0, 0, 0, 0, 0};
  __builtin_amdgcn_tensor_load_to_lds(g0, g1, g2, g3, g4, 0);
#else
  // ROCm 7.2 (clang-22): 5-arg builtin
  __builtin_amdgcn_tensor_load_to_lds(g0, g1, g2, g3, 0);
#endif
}
#else
#define HAVE_TDM 0
#endif

// ---------------------------------------------------------------------------
// Per-tile K-loop: D(16b x 16g) += A(16b x 4s) * B(4s x 16g) over one chunk.
// ---------------------------------------------------------------------------
__device__ __forceinline__ v8f wmma_chunk(const float* __restrict__ xt,
                                          const float* __restrict__ WtF,
                                          v8f acc, int m, int h, int lane) {
#pragma unroll 4
  for (int j = 0; j < KC / 4; ++j) {
    // A fragment: lane m holds row M=m; vgpr0/1 = K = 4j + 2h + {0,1}
    v2f a = *(const v2f*)(xt + m * KCP + 4 * j + 2 * h);
    // B fragment: lane holds col N=lane%16; vgpr0/1 = K = 4j + 2h + {0,1}
    v2f b = *(const v2f*)(&WtF[(j * 32 + lane) * 2]);
    acc = __builtin_amdgcn_wmma_f32_16x16x4_f32(
        false, a, false, b, (short)0, acc, false, false);
  }
  return acc;
}

// ---------------------------------------------------------------------------
// Kernel 2: block-diagonal GEMM via V_WMMA_F32_16X16X4_F32.
// Workgroup = 16 genes x 128 batch rows. Each wave owns 2 accumulator tiles.
// ---------------------------------------------------------------------------
__global__ __launch_bounds__(BLOCK_THREADS)
void gene_block_linear_kernel(const float* __restrict__ x,
                              const float* __restrict__ w,
                              const float* __restrict__ bias,
                              const int* __restrict__ gene_ids,
                              const int* __restrict__ seg_start,
                              float* __restrict__ out) {
  // Masked-weight tile, stored in WMMA B-fragment order: [j][lane][vgpr]
  __shared__ float WtF[(KC / 4) * 32 * 2];          // 4 KB
  __shared__ float Xs[WAVES * NTILE * 16 * KCP];    // ~34 KB

  const int tid  = threadIdx.x;
  const int wave = tid >> 5;
  const int lane = tid & 31;
  const int m = lane & 15;   // batch row within tile / gene column
  const int h = lane >> 4;   // half-wave select

  const int g0     = blockIdx.x * GENE_TILE;
  const int byBase = blockIdx.y * ROWS_PER_WG;

  const int s0   = seg_start[g0];
  const int sEnd = seg_start[g0 + GENE_TILE];

  v8f acc0, acc1;
#pragma unroll
  for (int i = 0; i < 8; ++i) { acc0[i] = 0.0f; acc1[i] = 0.0f; }

  const float* xt0 = &Xs[(wave * NTILE + 0) * 16 * KCP];
  const float* xt1 = &Xs[(wave * NTILE + 1) * 16 * KCP];

  for (int sc = s0; sc < sEnd; sc += KC) {
    // --- build masked weight tile Wt[k][n] = (gene_ids[s]==g0+n) ? w[s] : 0 ---
    for (int idx = tid; idx < KC * GENE_TILE; idx += BLOCK_THREADS) {
      int k = idx >> 4;            // SNP within chunk
      int n = idx & 15;            // gene column
      int s = sc + k;
      float v = 0.0f;
      if (s < S_DIM) v = (gene_ids[s] == g0 + n) ? w[s] : 0.0f;
      int j  = k >> 2;
      int kk = k & 3;
      int bl = n + ((kk >> 1) << 4);           // lane holding this element
      WtF[(j * 32 + bl) * 2 + (kk & 1)] = v;   // vgpr = kk&1
    }

    // --- stage x tiles (16 rows x KC cols each) into LDS ---
#if HAVE_TDM
    {
      int rowB0 = byBase + (wave * NTILE + 0) * 16;
      int rowB1 = byBase + (wave * NTILE + 1) * 16;
      tdm_load_x_tile((uint32_t)(uintptr_t)(const void*)xt0,
                      x + (size_t)rowB0 * S_DIM + sc, (unsigned)(S_DIM - sc));
      tdm_load_x_tile((uint32_t)(uintptr_t)(const void*)xt1,
                      x + (size_t)rowB1 * S_DIM + sc, (unsigned)(S_DIM - sc));
    }
#else
#pragma unroll
    for (int t = 0; t < NTILE; ++t) {
      int rowB = byBase + (wave * NTILE + t) * 16;
      float* xw = (float*)&Xs[(wave * NTILE + t) * 16 * KCP];
      for (int r = 0; r < 16; ++r) {
        int c = lane * 2;
        int s = sc + c;
        float v0 = (s < S_DIM) ? x[(size_t)(rowB + r) * S_DIM + s] : 0.0f;
        float v1 = (s + 1 < S_DIM) ? x[(size_t)(rowB + r) * S_DIM + s + 1] : 0.0f;
        xw[r * KCP + c]     = v0;
        xw[r * KCP + c + 1] = v1;
      }
    }
#endif
    __syncthreads();   // Wt visible to all waves

    // --- WMMA accumulation, tiles hand-unrolled so waits are compile-time ---
#if HAVE_TDM
    __builtin_amdgcn_s_wait_tensorcnt(1);   // tile 0's TDM done (in-order)
#endif
    acc0 = wmma_chunk(xt0, WtF, acc0, m, h, lane);
#if HAVE_TDM
    __builtin_amdgcn_s_wait_tensorcnt(0);   // tile 1's TDM done
#endif
    acc1 = wmma_chunk(xt1, WtF, acc1, m, h, lane);

    __syncthreads();   // all waves done reading Wt before next rebuild
  }

  // --- epilogue: bias + fused double-LeakyReLU, direct global store ---
  float bb = bias[g0 + m];
  {
    int rowBase0 = byBase + (wave * NTILE + 0) * 16 + h * 8;
    int rowBase1 = byBase + (wave * NTILE + 1) * 16 + h * 8;
#pragma unroll
    for (int i = 0; i < 8; ++i) {
      float z0 = acc0[i] + bb;               // C/D: vgpr i = row i (+8 for h=1), col = lane%16
      z0 = (z0 >= 0.0f) ? z0 : 0.02f * z0;   // leaky(leaky(z,0.1),0.2)
      out[(size_t)(rowBase0 + i) * G_DIM + (g0 + m)] = z0;
      float z1 = acc1[i] + bb;
      z1 = (z1 >= 0.0f) ? z1 : 0.02f * z1;
      out[(size_t)(rowBase1 + i) * G_DIM + (g0 + m)] = z1;
    }
  }
}

// ---------------------------------------------------------------------------
extern "C" void kernel_launch(void* const* d_in, const int* in_sizes, int n_in,
                              void* d_out, int out_size, void* d_ws, size_t ws_size,
                              hipStream_t stream) {
  (void)in_sizes; (void)n_in; (void)out_size; (void)ws_size;
  const float* x        = (const float*)d_in[0];
  const float* w        = (const float*)d_in[1];
  const float* bias     = (const float*)d_in[2];
  const int*   gene_ids = (const int*)d_in[3];
  int* seg_start = (int*)d_ws;   // (G_DIM+1) ints of scratch

  seg_offsets_kernel<<<dim3((G_DIM + 256) / 256), dim3(256), 0, stream>>>(
      gene_ids, seg_start);

  dim3 grid(G_DIM / GENE_TILE, NBY);
  gene_block_linear_kernel<<<grid, dim3(BLOCK_THREADS), 0, stream>>>(
      x, w, bias, gene_ids, seg_start, (float*)d_out);
}